// targetedLoss_6562710028353
// MI455X (gfx1250) — compile-verified
//
#include <hip/hip_runtime.h>
#include <math.h>

#define WDIM 512
#define LDIM 512
#define NPIX (WDIM * LDIM)      // 262144 pixels
#define MT   64                 // targets
#define BLOCK 256
#define NBLOCKS (NPIX / BLOCK)  // 1024 partial sums

typedef __attribute__((ext_vector_type(2))) float v2f;
typedef __attribute__((ext_vector_type(8))) float v8f;

// ---------------------------------------------------------------------------
// Main kernel: one thread per pixel, both anchors. Targets' standup 2D boxes
// (+ area) are computed once per block into LDS by threads 0..63.
// ---------------------------------------------------------------------------
__global__ __launch_bounds__(BLOCK)
void targeted_loss_main(const float* __restrict__ psm,     // [2, W, L]
                        const float* __restrict__ rm,      // [14, W, L]
                        const float* __restrict__ anc,     // [W, L, 2, 7]
                        const float* __restrict__ Tm,      // [4, 4] row-major
                        const float* __restrict__ tgt,     // [64, 7]
                        float* __restrict__ partial)       // [NBLOCKS]
{
    __shared__ float s_t[MT * 5];     // x1,y1,x2,y2,area per target
    __shared__ float s_red[BLOCK];

    const int tid = threadIdx.x;

    // ---- per-block target preprocessing (world-frame standup boxes) ----
    if (tid < MT) {
        const float* b = tgt + tid * 7;
        const float cx = b[0], cy = b[1];
        const float dh = b[3], dw = b[4], dl = b[5];   // boxes[:, [5,4,3]] = (l,w,h)
        const float yaw = b[6];
        const float c = cosf(yaw), s = sinf(yaw);
        const float hl = 0.5f * dl, hw = 0.5f * dw;
        (void)dh; // z extent does not affect 2D standup box
        float minx =  INFINITY, miny =  INFINITY;
        float maxx = -INFINITY, maxy = -INFINITY;
        #pragma unroll
        for (int k = 0; k < 4; ++k) {
            const float sx = (k & 1) ? -hl : hl;
            const float sy = (k & 2) ? -hw : hw;
            const float X = cx + (sx * c - sy * s);
            const float Y = cy + (sx * s + sy * c);
            minx = fminf(minx, X); maxx = fmaxf(maxx, X);
            miny = fminf(miny, Y); maxy = fmaxf(maxy, Y);
        }
        s_t[tid * 5 + 0] = minx;
        s_t[tid * 5 + 1] = miny;
        s_t[tid * 5 + 2] = maxx;
        s_t[tid * 5 + 3] = maxy;
        s_t[tid * 5 + 4] = (maxx - minx) * (maxy - miny);
    }
    __syncthreads();

    // transformation rows 0 and 1 (only x,y of projection are needed)
    const float T00 = Tm[0], T01 = Tm[1], T02 = Tm[2], T03 = Tm[3];
    const float T10 = Tm[4], T11 = Tm[5], T12 = Tm[6], T13 = Tm[7];

    const int pix = blockIdx.x * BLOCK + tid;
    float contrib = 0.0f;

    #pragma unroll
    for (int a = 0; a < 2; ++a) {
        // deltas: rm[(a*7+k), w, l] -- coalesced per channel
        float d[7];
        #pragma unroll
        for (int k = 0; k < 7; ++k)
            d[k] = rm[(a * 7 + k) * NPIX + pix];

        const float* an = anc + ((size_t)pix * 2 + a) * 7;
        const float ax = an[0], ay = an[1], az = an[2];
        const float ah = an[3], aw = an[4], al = an[5], ayaw = an[6];

        // box decode
        const float ad  = sqrtf(aw * aw + al * al);
        const float bx  = d[0] * ad + ax;
        const float by  = d[1] * ad + ay;
        const float bz  = d[2] * ah + az;
        const float bh  = expf(d[3]) * ah;
        const float bw  = expf(d[4]) * aw;
        const float bl  = expf(d[5]) * al;
        const float byaw = d[6] + ayaw;

        // 8 corners -> rotate -> transform -> 2D standup box
        const float c = cosf(byaw), s = sinf(byaw);
        const float hl = 0.5f * bl, hw = 0.5f * bw, hh = 0.5f * bh;
        float minx =  INFINITY, miny =  INFINITY;
        float maxx = -INFINITY, maxy = -INFINITY;
        #pragma unroll
        for (int k = 0; k < 8; ++k) {
            const float sx = (k & 1) ? -hl : hl;
            const float sy = (k & 2) ? -hw : hw;
            const float sz = (k & 4) ? -hh : hh;
            const float X = bx + (sx * c - sy * s);
            const float Y = by + (sx * s + sy * c);
            const float Z = bz + sz;
            const float px = T00 * X + T01 * Y + T02 * Z + T03;
            const float py = T10 * X + T11 * Y + T12 * Z + T13;
            minx = fminf(minx, px); maxx = fmaxf(maxx, px);
            miny = fminf(miny, py); maxy = fmaxf(maxy, py);
        }
        const float areaT = (maxx - minx) * (maxy - miny);

        // IoU sum against 64 targets (LDS broadcast reads)
        float siou = 0.0f;
        #pragma unroll 8
        for (int m = 0; m < MT; ++m) {
            const float tx1 = s_t[m * 5 + 0];
            const float ty1 = s_t[m * 5 + 1];
            const float tx2 = s_t[m * 5 + 2];
            const float ty2 = s_t[m * 5 + 3];
            const float ag  = s_t[m * 5 + 4];
            const float xx1 = fmaxf(minx, tx1);
            const float yy1 = fmaxf(miny, ty1);
            const float xx2 = fminf(maxx, tx2);
            const float yy2 = fminf(maxy, ty2);
            const float wv = fmaxf(0.0f, xx2 - xx1);
            const float hv = fmaxf(0.0f, yy2 - yy1);
            const float wh = wv * hv;
            siou += wh / (areaT + ag - wh);
        }

        // classification factor
        const float x    = psm[a * NPIX + pix];
        const float prob = 1.0f / (1.0f + expf(-x));
        if (prob > 0.1f)
            contrib += logf(1.0f - prob) * siou;
    }

    // ---- block tree reduction (deterministic) ----
    s_red[tid] = contrib;
    __syncthreads();
    #pragma unroll
    for (int off = BLOCK / 2; off > 0; off >>= 1) {
        if (tid < off) s_red[tid] += s_red[tid + off];
        __syncthreads();
    }
    if (tid == 0) partial[blockIdx.x] = s_red[0];
}

// ---------------------------------------------------------------------------
// Final reduction of NBLOCKS partials with V_WMMA_F32_16X16X4_F32.
// A = 16x4 tile of partials (64 values / instruction, 2 VGPRs), B = ones,
// D accumulates row-sums in fixed hardware order -> deterministic.
// Single wave32 block; EXEC is all-ones as WMMA requires.
// ---------------------------------------------------------------------------
__global__ __launch_bounds__(32)
void targeted_loss_reduce(const float* __restrict__ partial,
                          float* __restrict__ out)
{
    const int lane = threadIdx.x;   // 0..31
    v8f acc = {};
    v2f ones; ones.x = 1.0f; ones.y = 1.0f;

    #pragma unroll
    for (int it = 0; it < NBLOCKS / 64; ++it) {
        v2f av;
        av.x = partial[it * 64 + 2 * lane + 0];
        av.y = partial[it * 64 + 2 * lane + 1];
        // D = A(16x4) x ones(4x16) + C  => every column of D holds the 16 row sums
        acc = __builtin_amdgcn_wmma_f32_16x16x4_f32(
            /*neg_a=*/false, av, /*neg_b=*/false, ones,
            /*c_mod=*/(short)0, acc, /*reuse_a=*/false, /*reuse_b=*/false);
    }

    // lanes 0..15 hold rows 0..7 in v[0..7]; lanes 16..31 hold rows 8..15
    float s = acc[0] + acc[1] + acc[2] + acc[3] + acc[4] + acc[5] + acc[6] + acc[7];
    const float other = __shfl_xor(s, 16, 32);
    if (lane == 0) out[0] = s + other;
}

// ---------------------------------------------------------------------------
extern "C" void kernel_launch(void* const* d_in, const int* in_sizes, int n_in,
                              void* d_out, int out_size, void* d_ws, size_t ws_size,
                              hipStream_t stream)
{
    (void)in_sizes; (void)n_in; (void)out_size; (void)ws_size;
    const float* psm = (const float*)d_in[0];
    const float* rm  = (const float*)d_in[1];
    const float* anc = (const float*)d_in[2];
    const float* Tm  = (const float*)d_in[3];
    const float* tgt = (const float*)d_in[4];
    float* out = (float*)d_out;
    float* partial = (float*)d_ws;  // NBLOCKS floats = 4 KB

    targeted_loss_main<<<NBLOCKS, BLOCK, 0, stream>>>(psm, rm, anc, Tm, tgt, partial);
    targeted_loss_reduce<<<1, 32, 0, stream>>>(partial, out);
}